// SANL_2508260901618
// MI455X (gfx1250) — compile-verified
//
#include <hip/hip_runtime.h>
#include <hip/hip_bf16.h>

// ---------------------------------------------------------------------------
// CDNA5 (gfx1250) implementation of conv-conv-rank1-attention block.
// All matmul-shaped work runs on V_WMMA_F32_16X16X4_F32 (wave32 WMMA).
// ---------------------------------------------------------------------------

typedef __attribute__((ext_vector_type(2))) float v2f;
typedef __attribute__((ext_vector_type(8))) float v8f;

#define WMMA_F32X4(a, b, c) \
    __builtin_amdgcn_wmma_f32_16x16x4_f32(false, (a), false, (b), (short)0, (c), false, false)

#define L_FEAT 64
#define IMG_H 64
#define IMG_W 64
#define NBATCH 2
#define PLANE (IMG_H * IMG_W)           // 4096
#define NTOK (NBATCH * PLANE)           // 8192
#define FMAP_ELEMS (NBATCH * L_FEAT * PLANE)  // 524288 floats per feature map

// ---------------------------------------------------------------------------
// Kernel 1/2: 3x3 conv (64->64 chan) as per-tap implicit GEMM with WMMA.
// One block per (batch, image row). 8 waves; each wave owns two 16x16 C
// tiles that share the B fragment (input), differing only in A (weights).
// ---------------------------------------------------------------------------
__global__ __launch_bounds__(256) void conv3x3_wmma_kernel(
    const float* __restrict__ x,    // [B,64,64,64]
    const float* __restrict__ w,    // [64,64,3,3]
    const float* __restrict__ bias, // [64]
    float* __restrict__ y,          // [B,64,64,64]
    int relu)
{
    __shared__ float xs[3][L_FEAT][66];  // rows y-1..y+1, chan, col(-1..64)
    __shared__ float wt[L_FEAT][68];     // per-tap weights [oc][ic], padded

    const int tid = threadIdx.x;
    const int b   = blockIdx.x >> 6;
    const int row = blockIdx.x & 63;

    // Stage input halo tile (zero-padded borders), coalesced over columns.
    for (int i = tid; i < 3 * L_FEAT * 66; i += 256) {
        int r   = i / (L_FEAT * 66);
        int rem = i - r * (L_FEAT * 66);
        int c   = rem / 66;
        int col = rem - c * 66;
        int yy  = row - 1 + r;
        int xx  = col - 1;
        float val = 0.f;
        if (yy >= 0 && yy < IMG_H && xx >= 0 && xx < IMG_W)
            val = x[(((b * L_FEAT + c) * IMG_H + yy) * IMG_W) + xx];
        xs[r][c][col] = val;
    }

    const int wave = tid >> 5;
    const int lane = tid & 31;
    const int ml   = lane & 15;   // M (or N) index within a 16x16 tile
    const int kh   = lane >> 4;   // which K-half this lane supplies
    const int mt0  = wave >> 2;   // wave handles m-tiles mt0 and mt0+2
    const int nt   = wave & 3;    // shared n-tile

    v8f acc0 = {};
    v8f acc1 = {};

    for (int tap = 0; tap < 9; ++tap) {
        __syncthreads();  // xs ready / previous tap's wt reads done
        // wt[oc][ic] = w[oc][ic][tap]
        for (int i = tid; i < L_FEAT * L_FEAT; i += 256) {
            int oc = i >> 6, ic = i & 63;
            wt[oc][ic] = w[oc * 576 + ic * 9 + tap];
        }
        __syncthreads();

        const int dy   = tap / 3;
        const int dx   = tap - dy * 3;
        const int colb = nt * 16 + ml + dx;  // halo-based column index

        #pragma unroll
        for (int ks = 0; ks < 16; ++ks) {
            const int kb = ks * 4 + kh * 2;  // lane's 2 K values: kb, kb+1
            v2f a0 = *(const v2f*)&wt[mt0 * 16 + ml][kb];
            v2f a1 = *(const v2f*)&wt[(mt0 + 2) * 16 + ml][kb];
            v2f bf;
            bf.x = xs[dy][kb][colb];
            bf.y = xs[dy][kb + 1][colb];
            acc0 = WMMA_F32X4(a0, bf, acc0);
            acc1 = WMMA_F32X4(a1, bf, acc1);
        }
    }

    // C/D layout: VGPR r -> M = r + 8*kh, N = ml
    const int n = nt * 16 + ml;
    #pragma unroll
    for (int r = 0; r < 8; ++r) {
        int m0 = mt0 * 16 + r + 8 * kh;
        int m1 = (mt0 + 2) * 16 + r + 8 * kh;
        float v0 = acc0[r] + bias[m0];
        float v1 = acc1[r] + bias[m1];
        if (relu) { v0 = fmaxf(v0, 0.f); v1 = fmaxf(v1, 0.f); }
        y[(((b * L_FEAT + m0) * IMG_H + row) * IMG_W) + n] = v0;
        y[(((b * L_FEAT + m1) * IMG_H + row) * IMG_W) + n] = v1;
    }
}

// ---------------------------------------------------------------------------
// Kernel 3: q/k/v projections + L2 norm of q,k rows.
// Grid (b*row, matrix). Computes C[c_out x token] for one image row with
// WMMA, stages in LDS, normalizes per token, writes token-major [tok][64].
// ---------------------------------------------------------------------------
__global__ __launch_bounds__(256) void qkv_wmma_kernel(
    const float* __restrict__ h,   // [B,64,64,64] (conv2 output)
    const float* __restrict__ wq,
    const float* __restrict__ wk,
    const float* __restrict__ wv,
    float* __restrict__ qout,      // [8192,64] (L2-normalized rows)
    float* __restrict__ kout,      // [8192,64] (L2-normalized rows)
    float* __restrict__ vout)      // [8192,64]
{
    __shared__ float os[L_FEAT][68];  // [c_out][token], padded
    __shared__ float sc[L_FEAT];      // per-token inverse norm

    const int tid = threadIdx.x;
    const int b   = blockIdx.x >> 6;
    const int row = blockIdx.x & 63;
    const int mat = blockIdx.y;  // 0=q 1=k 2=v

    const float* wm = (mat == 0) ? wq : (mat == 1) ? wk : wv;
    float* om       = (mat == 0) ? qout : (mat == 1) ? kout : vout;

    const int wave = tid >> 5, lane = tid & 31;
    const int ml = lane & 15, kh = lane >> 4;
    const int mt0 = wave >> 2, nt = wave & 3;

    // h[b][d][row][x]: element (d, x) at hb[d*4096 + x]
    const float* hb = h + ((long)(b * L_FEAT) * IMG_H + row) * IMG_W;
    const int xcol  = nt * 16 + ml;

    v8f acc0 = {}, acc1 = {};
    #pragma unroll
    for (int ks = 0; ks < 16; ++ks) {
        const int kb = ks * 4 + kh * 2;
        v2f a0 = *(const v2f*)&wm[(mt0 * 16 + ml) * 64 + kb];        // W[c][d]
        v2f a1 = *(const v2f*)&wm[((mt0 + 2) * 16 + ml) * 64 + kb];
        v2f bf;
        bf.x = hb[kb * PLANE + xcol];
        bf.y = hb[(kb + 1) * PLANE + xcol];
        acc0 = WMMA_F32X4(a0, bf, acc0);
        acc1 = WMMA_F32X4(a1, bf, acc1);
    }

    const int n = nt * 16 + ml;
    #pragma unroll
    for (int r = 0; r < 8; ++r) {
        os[mt0 * 16 + r + 8 * kh][n]       = acc0[r];
        os[(mt0 + 2) * 16 + r + 8 * kh][n] = acc1[r];
    }
    __syncthreads();

    if (tid < 64) {
        if (mat < 2) {
            float s = 0.f;
            #pragma unroll
            for (int c = 0; c < 64; ++c) { float t = os[c][tid]; s += t * t; }
            sc[tid] = 1.f / fmaxf(sqrtf(s), 1e-12f);
        } else {
            sc[tid] = 1.f;
        }
    }
    __syncthreads();

    const long g0 = (long)b * PLANE + (long)row * IMG_W;  // first token of row
    for (int i = tid; i < L_FEAT * IMG_W; i += 256) {
        int tok = i >> 6, c = i & 63;
        om[(g0 + tok) * 64 + c] = os[c][tok] * sc[tok];
    }
}

// ---------------------------------------------------------------------------
// Kernel 4: per-token rank-1 softmax attention. One wave32 per token.
// scores row c = q_c * k[:]; softmax over d folds to a scalar-scaled k,
// with row-max = q_c * (q_c>=0 ? max(k) : min(k)).
// ---------------------------------------------------------------------------
__global__ __launch_bounds__(256) void rank1_attn_kernel(
    const float* __restrict__ q,   // [8192,64]
    const float* __restrict__ k,
    const float* __restrict__ v,
    float* __restrict__ att)       // [8192,64]
{
    const int lane = threadIdx.x & 31;
    const int wv_  = threadIdx.x >> 5;
    const int tok  = blockIdx.x * 8 + wv_;

    const float* qp = q + (long)tok * 64;
    const float* kp = k + (long)tok * 64;
    const float* vp = v + (long)tok * 64;

    float k0 = kp[lane], k1 = kp[lane + 32];
    float v0 = vp[lane], v1 = vp[lane + 32];
    float q0 = qp[lane], q1 = qp[lane + 32];

    float mx = fmaxf(k0, k1), mn = fminf(k0, k1);
    #pragma unroll
    for (int off = 16; off > 0; off >>= 1) {
        mx = fmaxf(mx, __shfl_xor(mx, off, 32));
        mn = fminf(mn, __shfl_xor(mn, off, 32));
    }

    float* ap = att + (long)tok * 64;
    #pragma unroll 4
    for (int c = 0; c < 64; ++c) {
        float qc = (c < 32) ? __shfl(q0, c, 32) : __shfl(q1, c - 32, 32);
        float m  = qc * ((qc >= 0.f) ? mx : mn);
        float e0 = __expf(qc * k0 - m);
        float e1 = __expf(qc * k1 - m);
        float s   = e0 + e1;
        float num = e0 * v0 + e1 * v1;
        #pragma unroll
        for (int off = 16; off > 0; off >>= 1) {
            s   += __shfl_xor(s, off, 32);
            num += __shfl_xor(num, off, 32);
        }
        if (lane == 0) ap[c] = num / s;
    }
}

// ---------------------------------------------------------------------------
// Kernel 5: out = att @ wo^T + bo, transpose to NCHW, add conv2 residual.
// C[c2 x token] per image row; A (wo) and B (att) are direct global float2
// loads (both K-pairs contiguous). Stores are fully coalesced NCHW.
// ---------------------------------------------------------------------------
__global__ __launch_bounds__(256) void proj_residual_kernel(
    const float* __restrict__ att,  // [8192,64]
    const float* __restrict__ wo,   // [64,64]
    const float* __restrict__ bo,   // [64]
    const float* __restrict__ h,    // [B,64,64,64] residual
    float* __restrict__ out)        // [B,64,64,64]
{
    const int tid = threadIdx.x;
    const int b   = blockIdx.x >> 6;
    const int row = blockIdx.x & 63;
    const int wave = tid >> 5, lane = tid & 31;
    const int ml = lane & 15, kh = lane >> 4;
    const int mt0 = wave >> 2, nt = wave & 3;

    const float* ab = att + ((long)b * PLANE + (long)row * IMG_W) * 64;
    const int tokn  = nt * 16 + ml;

    v8f acc0 = {}, acc1 = {};
    #pragma unroll
    for (int ks = 0; ks < 16; ++ks) {
        const int kb = ks * 4 + kh * 2;
        v2f a0 = *(const v2f*)&wo[(mt0 * 16 + ml) * 64 + kb];
        v2f a1 = *(const v2f*)&wo[((mt0 + 2) * 16 + ml) * 64 + kb];
        v2f bf = *(const v2f*)&ab[tokn * 64 + kb];
        acc0 = WMMA_F32X4(a0, bf, acc0);
        acc1 = WMMA_F32X4(a1, bf, acc1);
    }

    const int x = nt * 16 + ml;
    #pragma unroll
    for (int r = 0; r < 8; ++r) {
        int c0 = mt0 * 16 + r + 8 * kh;
        int c1 = (mt0 + 2) * 16 + r + 8 * kh;
        long i0 = (((long)(b * L_FEAT + c0) * IMG_H + row) * IMG_W) + x;
        long i1 = (((long)(b * L_FEAT + c1) * IMG_H + row) * IMG_W) + x;
        out[i0] = acc0[r] + bo[c0] + h[i0];
        out[i1] = acc1[r] + bo[c1] + h[i1];
    }
}

// ---------------------------------------------------------------------------
extern "C" void kernel_launch(void* const* d_in, const int* in_sizes, int n_in,
                              void* d_out, int out_size, void* d_ws, size_t ws_size,
                              hipStream_t stream) {
    (void)in_sizes; (void)n_in; (void)out_size; (void)ws_size;

    const float* x    = (const float*)d_in[0];
    const float* c1w  = (const float*)d_in[1];
    const float* c1b  = (const float*)d_in[2];
    const float* c2w  = (const float*)d_in[3];
    const float* c2b  = (const float*)d_in[4];
    const float* wq   = (const float*)d_in[5];
    const float* wk   = (const float*)d_in[6];
    const float* wv   = (const float*)d_in[7];
    const float* wo   = (const float*)d_in[8];
    const float* bo   = (const float*)d_in[9];

    float* ws  = (float*)d_ws;
    float* h1  = ws;                   // [B,64,64,64]
    float* h2  = ws + 1 * FMAP_ELEMS;  // [B,64,64,64]
    float* q   = ws + 2 * FMAP_ELEMS;  // [8192,64]
    float* k   = ws + 3 * FMAP_ELEMS;  // [8192,64]
    float* v   = ws + 4 * FMAP_ELEMS;  // [8192,64]
    float* att = ws + 5 * FMAP_ELEMS;  // [8192,64]

    const int rows = NBATCH * IMG_H;   // 128 blocks: one per image row

    conv3x3_wmma_kernel<<<rows, 256, 0, stream>>>(x, c1w, c1b, h1, 1);
    conv3x3_wmma_kernel<<<rows, 256, 0, stream>>>(h1, c2w, c2b, h2, 0);
    qkv_wmma_kernel<<<dim3(rows, 3), 256, 0, stream>>>(h2, wq, wk, wv, q, k, v);
    rank1_attn_kernel<<<NTOK / 8, 256, 0, stream>>>(q, k, v, att);
    proj_residual_kernel<<<rows, 256, 0, stream>>>(att, wo, bo, h2, (float*)d_out);
}